// Embedding_79053168050815
// MI455X (gfx1250) — compile-verified
//
#include <hip/hip_runtime.h>
#include <stdint.h>

// Embedding gather out[r,:] = W[x[r],:] implemented entirely with the CDNA5
// Tensor Data Mover:
//   1) TENSOR_LOAD_TO_LDS in gather mode pulls 8 indexed 4KB rows of W to LDS
//   2) s_wait_tensorcnt 0
//   3) TENSOR_STORE_FROM_LDS writes the contiguous 8x1024 f32 tile to out
// The wave's only job is building two descriptors; all data movement is DMA.

typedef __attribute__((ext_vector_type(4))) unsigned int v4u;
typedef __attribute__((ext_vector_type(4))) int          v4i;
typedef __attribute__((ext_vector_type(8))) int          v8i;

#define EMBED        1024
#define VOCAB        50257
#define ROWS_PER_BLK 8

__global__ __launch_bounds__(32) void embed_gather_tdm(
    const int* __restrict__ x, const float* __restrict__ W,
    float* __restrict__ out, int total_rows)
{
    __shared__ float lds[ROWS_PER_BLK * EMBED];   // 32 KB staging tile

    const int base = blockIdx.x * ROWS_PER_BLK;
    int rows_here  = total_rows - base;
    if (rows_here > ROWS_PER_BLK) rows_here = ROWS_PER_BLK;

    // LDS byte offset of the staging tile: low 32 bits of the generic address
    // (generic LDS addr = {SHARED_BASE,16'b0} in [63:32], LDS offset in [31:0]).
    // Also keeps the shared array referenced -> 32 KB group segment allocated.
    const unsigned int lds_base = (unsigned int)(uintptr_t)(&lds[0]);

    // Uniform (blockIdx-only) index loads -> scalar loads. Clamp for the tail;
    // slots >= rows_here are ignored (tile_dim1 = rows_here).
    const int last = total_rows - 1;
    int i0 = x[min(base + 0, last)];
    int i1 = x[min(base + 1, last)];
    int i2 = x[min(base + 2, last)];
    int i3 = x[min(base + 3, last)];
    int i4 = x[min(base + 4, last)];
    int i5 = x[min(base + 5, last)];
    int i6 = x[min(base + 6, last)];
    int i7 = x[min(base + 7, last)];

    // ---------------- Descriptor 1: gather load W rows -> LDS ----------------
    const uint64_t wa = (uint64_t)(uintptr_t)W;

    v4u ld0;
    // count=1, gather_index_size=1 (32-bit indices), gather_mode=1
    ld0.x = 1u | (1u << 30) | (1u << 31);
    ld0.y = lds_base;                                    // lds_addr (bytes)
    ld0.z = (unsigned int)(wa & 0xFFFFFFFFu);            // global_addr[31:0]
    ld0.w = (unsigned int)((wa >> 32) & 0x01FFFFFFu) | (2u << 30); // [56:32]|type=2

    v8i ld1;
    ld1[0] = (int)(2u << 16);                            // data_size=2 (4 bytes)
    // tensor_dim0=EMBED [79:48], tensor_dim1=VOCAB [111:80], tile_dim0=EMBED [127:112]
    ld1[1] = (int)(((unsigned)EMBED & 0xFFFFu) << 16);
    ld1[2] = (int)((((unsigned)EMBED >> 16) & 0xFFFFu) |
                   (((unsigned)VOCAB & 0xFFFFu) << 16));
    ld1[3] = (int)((((unsigned)VOCAB >> 16) & 0xFFFFu) |
                   (((unsigned)EMBED & 0xFFFFu) << 16));
    ld1[4] = rows_here & 0xFFFF;                         // tile_dim1 = #indices
    ld1[5] = EMBED;                                      // tensor_dim0_stride lo32
    ld1[6] = 0;
    ld1[7] = 0;

    v4i ld2; ld2[0] = i0; ld2[1] = i1; ld2[2] = i2; ld2[3] = i3;  // row idx 0..3
    v4i ld3; ld3[0] = i4; ld3[1] = i5; ld3[2] = i6; ld3[3] = i7;  // row idx 4..7

    // ---------------- Descriptor 2: store LDS tile -> out (2D) ---------------
    const uint64_t oa = (uint64_t)(uintptr_t)(out + (size_t)base * EMBED);

    v4u st0;
    st0.x = 1u;                                          // count=1, normal mode
    st0.y = lds_base;                                    // lds_addr (bytes)
    st0.z = (unsigned int)(oa & 0xFFFFFFFFu);
    st0.w = (unsigned int)((oa >> 32) & 0x01FFFFFFu) | (2u << 30);

    v8i st1;
    st1[0] = (int)(2u << 16);                            // data_size=2 (4 bytes)
    // tensor_dim0=EMBED, tensor_dim1=rows_here, tile_dim0=EMBED
    st1[1] = (int)(((unsigned)EMBED & 0xFFFFu) << 16);
    st1[2] = (int)((((unsigned)EMBED >> 16) & 0xFFFFu) |
                   (((unsigned)rows_here & 0xFFFFu) << 16));
    st1[3] = (int)(((unsigned)EMBED & 0xFFFFu) << 16);   // rows_here>>16==0
    st1[4] = rows_here & 0xFFFF;                         // tile_dim1 (tile_dim2=0)
    st1[5] = EMBED;                                      // tensor_dim0_stride lo32
    st1[6] = 0;
    st1[7] = 0;

    v4i z4; z4[0] = 0; z4[1] = 0; z4[2] = 0; z4[3] = 0;  // 2D: groups 2/3 unused

#if defined(__clang_major__) && (__clang_major__ >= 23)
    v8i z8; z8[0]=0; z8[1]=0; z8[2]=0; z8[3]=0; z8[4]=0; z8[5]=0; z8[6]=0; z8[7]=0;
    __builtin_amdgcn_tensor_load_to_lds(ld0, ld1, ld2, ld3, z8, 0);
    __builtin_amdgcn_s_wait_tensorcnt(0);                // LDS tile is ready
    __builtin_amdgcn_tensor_store_from_lds(st0, st1, z4, z4, z8, 0);
#else
    __builtin_amdgcn_tensor_load_to_lds(ld0, ld1, ld2, ld3, 0);
    __builtin_amdgcn_s_wait_tensorcnt(0);                // LDS tile is ready
    __builtin_amdgcn_tensor_store_from_lds(st0, st1, z4, z4, 0);
#endif

    // Ensure the store DMA completed before the wave retires (S_ENDPGM also
    // performs an implicit wait-idle; this is explicit and cheap).
    __builtin_amdgcn_s_wait_tensorcnt(0);
}

extern "C" void kernel_launch(void* const* d_in, const int* in_sizes, int n_in,
                              void* d_out, int out_size, void* d_ws, size_t ws_size,
                              hipStream_t stream) {
    const int*   x = (const int*)d_in[0];    // [8, 2048] token ids
    const float* W = (const float*)d_in[1];  // [50257, 1024] f32
    float*     out = (float*)d_out;          // [8, 2048, 1024] f32

    const int total_rows = in_sizes[0];      // 16384
    const int blocks = (total_rows + ROWS_PER_BLK - 1) / ROWS_PER_BLK;

    embed_gather_tdm<<<blocks, 32, 0, stream>>>(x, W, out, total_rows);
}